// MarkedPointProcessRMTPPModel_27204322853650
// MI455X (gfx1250) — compile-verified
//
#include <hip/hip_runtime.h>
#include <cstdint>

// ---------------------------------------------------------------------------
// RMTPP intensity-over-grid for MI455X (gfx1250, wave32, WMMA).
//
// Pass 1: durations[b] = max_t ts[t,b]                       (tiny)
// Pass 2: proj[t,b]    = dot(h[t,b,:], w)  via V_WMMA_F32_16X16X4_F32
//         -> bandwidth bound: reads T*BS*H*4 = 256 MB  (~11 us @ 23.3 TB/s)
// Pass 3: per (n,b): grid = n/(N-1)*dur[b]; idx = first t with ts[t,b]<grid
//         (argmax-of-bool semantics; early-exit scan -> 1 iter on sorted ts);
//         log = proj[idx,b] + ti*(grid - ts[idx,b]) + bi     (1 MB written)
// ---------------------------------------------------------------------------

typedef float v2f __attribute__((ext_vector_type(2)));
typedef float v8f __attribute__((ext_vector_type(8)));

__global__ void rmtpp_durations_kernel(const float* __restrict__ ts,
                                       float* __restrict__ dur,
                                       int T, int BS) {
    int b = blockIdx.x * blockDim.x + threadIdx.x;
    if (b >= BS) return;
    float m = -3.402823466e38f;
    for (int t = 0; t < T; ++t) {
        m = fmaxf(m, ts[(size_t)t * BS + b]);   // coalesced across b
    }
    dur[b] = m;
}

// proj[row] = dot(h[row, :], w) for 16 rows per wave, via f32 WMMA.
// A (16x4 f32): lane L<16 holds A[L][k],A[L][k+1]; lane L+16 holds A[L][k+2..3]
// B (4x16 f32): broadcast w chunk -> every column identical
// D (16x16)   : D[m][n] = proj[rowBase+m] for all n.
__global__ void rmtpp_proj_wmma_kernel(const float* __restrict__ h,
                                       const float* __restrict__ w,
                                       float* __restrict__ proj,
                                       int rows, int H) {
    const int lane        = threadIdx.x & 31;
    const int waveInBlock = threadIdx.x >> 5;
    const int wave        = blockIdx.x * (blockDim.x >> 5) + waveInBlock;
    const int rowBase     = wave * 16;
    if (rowBase >= rows) return;              // wave-uniform: EXEC stays all-1s

    const int m    = lane & 15;
    const int koff = (lane >> 4) << 1;        // 0 for lanes 0-15, 2 for 16-31
    int r = rowBase + m;
    if (r >= rows) r = rows - 1;              // clamp tail (store-masked below)

    const float* __restrict__ rowPtr = h + (size_t)r * H + koff;
    const float* __restrict__ wPtr   = w + koff;

    v8f c = {};
    #pragma unroll 4
    for (int k = 0; k < H; k += 4) {
        v2f a = *reinterpret_cast<const v2f*>(rowPtr + k);  // 8B/lane, aligned
        v2f b = *reinterpret_cast<const v2f*>(wPtr + k);    // L0-resident
        // 8 args: (neg_a, A, neg_b, B, c_mod, C, reuse_a, reuse_b)
        c = __builtin_amdgcn_wmma_f32_16x16x4_f32(
                false, a, false, b, (short)0, c, false, false);
    }

    // Extract one value per output row from the 16x16 D tile:
    //   VGPR rIdx: lanes 0-15 hold M=rIdx, lanes 16-31 hold M=rIdx+8.
    const int sel = lane & 7;
    float v = c[0];
    v = (sel == 1) ? c[1] : v;
    v = (sel == 2) ? c[2] : v;
    v = (sel == 3) ? c[3] : v;
    v = (sel == 4) ? c[4] : v;
    v = (sel == 5) ? c[5] : v;
    v = (sel == 6) ? c[6] : v;
    v = (sel == 7) ? c[7] : v;

    const int  mOut    = ((lane >> 4) << 3) | sel;  // lanes 0-7 -> m, 16-23 -> 8+m
    const bool doStore = (lane < 8) || (lane >= 16 && lane < 24);
    if (doStore && (rowBase + mOut) < rows) {
        proj[rowBase + mOut] = v;
    }
}

__global__ void rmtpp_grid_kernel(const float* __restrict__ ts,
                                  const float* __restrict__ proj,
                                  const float* __restrict__ dur,
                                  const float* __restrict__ time_influence,
                                  const float* __restrict__ base_intensity,
                                  float* __restrict__ out_log,
                                  float* __restrict__ out_grid,
                                  int T, int BS, int NG) {
    const int gid = blockIdx.x * blockDim.x + threadIdx.x;
    if (gid >= NG * BS) return;
    const int b = gid % BS;         // b fastest -> coalesced stores
    const int n = gid / BS;

    const float ti = time_influence[0];   // scalar (s_load)
    const float bi = base_intensity[0];

    const float t_grid = ((float)n / (float)(NG - 1)) * dur[b];

    // jnp.argmax(ts < t_grid, axis=0): index of FIRST True; 0 if none.
    int idx = 0;
    for (int t = 0; t < T; ++t) {
        if (ts[(size_t)t * BS + b] < t_grid) { idx = t; break; }
    }

    const float closest_ts = ts[(size_t)idx * BS + b];
    const float li = proj[(size_t)idx * BS + b] + ti * (t_grid - closest_ts) + bi;

    out_log[gid]  = li;
    out_grid[gid] = t_grid;
}

extern "C" void kernel_launch(void* const* d_in, const int* in_sizes, int n_in,
                              void* d_out, int out_size, void* d_ws, size_t ws_size,
                              hipStream_t stream) {
    const float* h  = (const float*)d_in[0];   // (T, BS, H)
    const float* ts = (const float*)d_in[1];   // (T, BS, 1)
    const float* w  = (const float*)d_in[2];   // (H, 1)
    const float* ti = (const float*)d_in[3];   // (1,)
    const float* bi = (const float*)d_in[4];   // (1,)

    const int NG  = 1000;                      // hard-coded in reference
    const int BS  = out_size / (2 * NG);       // out = log(N*BS) ++ grid(N*BS)
    const int TBS = in_sizes[1];               // T*BS
    const int T   = TBS / BS;
    const int H   = in_sizes[2];

    float* out_log  = (float*)d_out;
    float* out_grid = out_log + (size_t)NG * BS;

    float* proj = (float*)d_ws;                // T*BS floats (512 KB)
    float* dur  = proj + TBS;                  // BS floats

    // Pass 1: per-sequence max timestamp
    rmtpp_durations_kernel<<<(BS + 127) / 128, 128, 0, stream>>>(ts, dur, T, BS);

    // Pass 2: GEMV projection via WMMA. 16 rows/wave, 8 waves/block.
    const int waves  = (TBS + 15) / 16;
    const int blocks = (waves + 7) / 8;
    rmtpp_proj_wmma_kernel<<<blocks, 256, 0, stream>>>(h, w, proj, TBS, H);

    // Pass 3: grid evaluation + gather
    const int total = NG * BS;
    rmtpp_grid_kernel<<<(total + 255) / 256, 256, 0, stream>>>(
        ts, proj, dur, ti, bi, out_log, out_grid, T, BS, NG);
}